// MultiHeadGAT_12017318494743
// MI455X (gfx1250) — compile-verified
//
#include <hip/hip_runtime.h>
#include <hip/hip_bf16.h>
#include <stdint.h>

typedef float v2f __attribute__((ext_vector_type(2)));
typedef float v8f __attribute__((ext_vector_type(8)));

#define HEADS 8
#define HID 64
#define NEG_SLOPE 0.2f
#define BN_EPS 1e-5f
#define SM_EPS 1e-16f

// ---------- helpers ----------
__device__ __forceinline__ float leaky(float x) { return x > 0.f ? x : NEG_SLOPE * x; }

// monotonic float -> uint encoding for atomic max over signed floats
__device__ __forceinline__ unsigned fenc(float x) {
    unsigned b = __float_as_uint(x);
    return (b & 0x80000000u) ? ~b : (b | 0x80000000u);
}
__device__ __forceinline__ float fdec(unsigned e) {
    unsigned b = (e & 0x80000000u) ? (e & 0x7FFFFFFFu) : ~e;
    return __uint_as_float(b);
}

__device__ __forceinline__ void edge_sd(const long long* __restrict__ ei, long e,
                                        int n_edges, long* s, long* d) {
    if (e < (long)n_edges) { *s = (long)ei[e]; *d = (long)ei[(long)n_edges + e]; }
    else                   { *s = *d = e - n_edges; }   // PyG add_self_loops
}

// ---------- dense transform: H[head][tile of 16 nodes][64] = X_tile @ W[head] ----------
// 256-thread block = 8 waves, each wave owns one 16-node tile of the same head.
// W[head] (K x 64) is staged into LDS via the gfx1250 async global->LDS copy
// (ASYNCcnt), then re-laid-out pair-interleaved so each WMMA B fragment is a
// single contiguous ds_load_b64. Each wave runs f32 WMMA 16x16x4.
__global__ void __launch_bounds__(256)
gat_wmma_gemm(const float* __restrict__ in, long in_head_stride, int K,
              const float* __restrict__ W,   // [HEADS][K][HID]
              float* __restrict__ out,       // [HEADS][N][HID]
              int n_nodes) {
    __shared__ __align__(16) float Wraw[128 * HID];   // raw K x 64 tile (<=32KB)
    __shared__ __align__(16) float Wint[128 * HID];   // pair-interleaved copy
    const int head = blockIdx.y;
    const float* __restrict__ Wh = W + (long)head * K * HID;
    const int nelem = K * HID;

    // ---- async stage W[head] into LDS: 16B per lane per issue ----
    {
        const unsigned ldsbase = (unsigned)(uintptr_t)(void*)Wraw;  // low 32 bits = LDS offset
        const int nbytes = nelem * (int)sizeof(float);
        for (int off = threadIdx.x * 16; off < nbytes; off += blockDim.x * 16) {
            const unsigned laddr = ldsbase + (unsigned)off;
            const char* gp = (const char*)Wh + off;
            asm volatile("global_load_async_to_lds_b128 %0, %1, off"
                         :: "v"(laddr), "v"(gp) : "memory");
        }
        asm volatile("s_wait_asynccnt 0x0" ::: "memory");
        __syncthreads();
    }
    // ---- interleave: Wint[(k/2)*128 + n*2 + (k&1)] = W[k][n] ----
    for (int idx = threadIdx.x; idx < nelem; idx += blockDim.x) {
        const int k = idx >> 6;          // idx / HID
        const int n = idx & 63;          // idx % HID
        Wint[(k >> 1) * 128 + n * 2 + (k & 1)] = Wraw[idx];
    }
    __syncthreads();

    const int wave = threadIdx.x >> 5;
    const int lane = threadIdx.x & 31;
    const int tile = blockIdx.x * 8 + wave;
    const int row0 = tile * 16;
    if (row0 < n_nodes) {                              // wave-uniform
        const int r    = lane & 15;
        const int half = lane >> 4;                    // K-half select (ISA 7.12.2 f32 layout)
        int node = row0 + r;
        if (node >= n_nodes) node = n_nodes - 1;       // clamp; garbage rows never stored
        const float* __restrict__ X = in + (long)head * in_head_stride + (long)node * K;

        v8f a0 = {}, a1 = {}, a2 = {}, a3 = {};
        for (int k0 = 0; k0 < K; k0 += 4) {
            const int kA = k0 + half * 2;
            const v2f av = *(const v2f*)(X + kA);      // global_load_b64, unpredicated
            // pair index kA/2 = k0/2 + half; fragment t at columns t*16 + r
            const float* wp = Wint + (k0 >> 1) * 128 + half * 128 + r * 2;
            const v2f b0 = *(const v2f*)(wp);          // ds_load_b64 each
            const v2f b1 = *(const v2f*)(wp + 32);
            const v2f b2 = *(const v2f*)(wp + 64);
            const v2f b3 = *(const v2f*)(wp + 96);
            a0 = __builtin_amdgcn_wmma_f32_16x16x4_f32(false, av, false, b0, (short)0, a0, false, false);
            a1 = __builtin_amdgcn_wmma_f32_16x16x4_f32(false, av, false, b1, (short)0, a1, false, false);
            a2 = __builtin_amdgcn_wmma_f32_16x16x4_f32(false, av, false, b2, (short)0, a2, false, false);
            a3 = __builtin_amdgcn_wmma_f32_16x16x4_f32(false, av, false, b3, (short)0, a3, false, false);
        }
        // C/D layout: VGPR i -> row (half*8+i), col = lane%16 (per 16-wide N tile)
        for (int i = 0; i < 8; ++i) {
            const int nd = row0 + half * 8 + i;
            if (nd < n_nodes) {
                float* o = out + ((long)head * n_nodes + nd) * HID;
                o[r]      = a0[i];
                o[r + 16] = a1[i];
                o[r + 32] = a2[i];
                o[r + 48] = a3[i];
            }
        }
    }
}

// ---------- per-node attention scores: s_src = h . a_src, s_dst = h . a_dst ----------
__global__ void attn_scores(const float* __restrict__ h, const float* __restrict__ a_src,
                            const float* __restrict__ a_dst,
                            float* __restrict__ s_src, float* __restrict__ s_dst, int n_nodes) {
    long t = (long)blockIdx.x * blockDim.x + threadIdx.x;      // over HEADS*N
    const long total = (long)HEADS * n_nodes;
    if (t >= total) return;
    const int head = (int)(t / n_nodes);
    const float* hr = h + t * HID;
    const float* as = a_src + head * HID;
    const float* ad = a_dst + head * HID;
    float ss = 0.f, sd = 0.f;
    #pragma unroll 8
    for (int c = 0; c < HID; ++c) { const float v = hr[c]; ss += v * as[c]; sd += v * ad[c]; }
    s_src[t] = ss;
    s_dst[t] = sd;
}

// ---------- edge pass 1: segment max of leaky-relu logits over dst ----------
__global__ void edge_max(const long long* __restrict__ ei,
                         const float* __restrict__ s_src, const float* __restrict__ s_dst,
                         unsigned* __restrict__ menc, int n_nodes, int n_edges, int e_tot) {
    long t = (long)blockIdx.x * blockDim.x + threadIdx.x;
    if (t >= (long)e_tot * HEADS) return;
    const int hh = (int)(t & (HEADS - 1));
    const long e = t >> 3;
    long s, d; edge_sd(ei, e, n_edges, &s, &d);
    const long base = (long)hh * n_nodes;
    const float lg = leaky(s_src[base + s] + s_dst[base + d]);
    atomicMax(&menc[base + d], fenc(lg));
}

// ---------- edge pass 2: denom = segment sum of exp(logit - max) ----------
__global__ void edge_denom(const long long* __restrict__ ei,
                           const float* __restrict__ s_src, const float* __restrict__ s_dst,
                           const unsigned* __restrict__ menc, float* __restrict__ denom,
                           int n_nodes, int n_edges, int e_tot) {
    long t = (long)blockIdx.x * blockDim.x + threadIdx.x;
    if (t >= (long)e_tot * HEADS) return;
    const int hh = (int)(t & (HEADS - 1));
    const long e = t >> 3;
    long s, d; edge_sd(ei, e, n_edges, &s, &d);
    const long base = (long)hh * n_nodes;
    const float lg = leaky(s_src[base + s] + s_dst[base + d]);
    const float ex = expf(lg - fdec(menc[base + d]));
    atomicAdd(&denom[base + d], ex);
}

// ---------- edge pass 3: agg[dst] += alpha * h[src]; one wave32 per edge ----------
__global__ void edge_aggregate(const long long* __restrict__ ei,
                               const float* __restrict__ s_src, const float* __restrict__ s_dst,
                               const unsigned* __restrict__ menc, const float* __restrict__ denom,
                               const float* __restrict__ h, float* __restrict__ agg,
                               int n_nodes, int n_edges, int e_tot) {
    const int wave = blockIdx.x * (blockDim.x >> 5) + (threadIdx.x >> 5);
    const int lane = threadIdx.x & 31;
    if (wave >= e_tot) return;
    long s, d; edge_sd(ei, (long)wave, n_edges, &s, &d);
    for (int hh = 0; hh < HEADS; ++hh) {
        const long base = (long)hh * n_nodes;
        const float lg    = leaky(s_src[base + s] + s_dst[base + d]);     // uniform across wave
        const float alpha = expf(lg - fdec(menc[base + d])) / (denom[base + d] + SM_EPS);
        const float* hs = h   + (base + s) * HID;
        float*       od = agg + (base + d) * HID;
        atomicAdd(&od[lane],      alpha * hs[lane]);
        atomicAdd(&od[lane + 32], alpha * hs[lane + 32]);
    }
}

// ---------- BN statistics: per (head, channel) sum & sumsq of (agg + bias) ----------
__global__ void bn_stats(const float* __restrict__ agg, const float* __restrict__ bias,
                         float* __restrict__ sums, float* __restrict__ sumsq, int n_nodes) {
    const int head = blockIdx.y;
    const int c    = threadIdx.x & 63;
    const int sub  = threadIdx.x >> 6;               // 0..3
    __shared__ float ls[4][HID];
    __shared__ float lq[4][HID];
    const float b = bias[head * HID + c];
    float s = 0.f, q = 0.f;
    for (int node = blockIdx.x * 4 + sub; node < n_nodes; node += gridDim.x * 4) {
        const float v = agg[((long)head * n_nodes + node) * HID + c] + b;
        s += v; q += v * v;
    }
    ls[sub][c] = s; lq[sub][c] = q;
    __syncthreads();
    if (sub == 0) {
        s = ls[0][c] + ls[1][c] + ls[2][c] + ls[3][c];
        q = lq[0][c] + lq[1][c] + lq[2][c] + lq[3][c];
        atomicAdd(&sums[head * HID + c], s);
        atomicAdd(&sumsq[head * HID + c], q);
    }
}

// ---------- fused bias + BN(train-mode batch stats) + ELU, in-place ----------
__global__ void bn_apply_elu(float* __restrict__ agg, const float* __restrict__ bias,
                             const float* __restrict__ sums, const float* __restrict__ sumsq,
                             const float* __restrict__ gamma, const float* __restrict__ beta,
                             int n_nodes) {
    long t = (long)blockIdx.x * blockDim.x + threadIdx.x;   // over HEADS*N*HID
    const long total = (long)HEADS * n_nodes * HID;
    if (t >= total) return;
    const int c  = (int)(t & (HID - 1));
    const long nh = t >> 6;                                  // head*N + node
    const int head = (int)(nh / n_nodes);
    const int hc = head * HID + c;
    const float invn = 1.f / (float)n_nodes;
    const float mu  = sums[hc] * invn;
    const float var = sumsq[hc] * invn - mu * mu;
    const float v = agg[t] + bias[hc];
    const float y = (v - mu) * rsqrtf(var + BN_EPS) * gamma[hc] + beta[hc];
    agg[t] = y > 0.f ? y : (expf(y) - 1.f);                  // ELU(alpha=1)
}

// ---------- ensemble classifier: out[n, cls] = concat_heads(h) @ Wc + bc ----------
__global__ void classify(const float* __restrict__ hfin, const float* __restrict__ Wc,
                         const float* __restrict__ bc, float* __restrict__ out, int n_nodes) {
    const int node = blockIdx.x * blockDim.x + threadIdx.x;
    if (node >= n_nodes) return;
    float o0 = bc[0], o1 = bc[1];
    for (int head = 0; head < HEADS; ++head) {
        const float* hr = hfin + ((long)head * n_nodes + node) * HID;
        const float* w  = Wc + head * HID * 2;
        #pragma unroll 8
        for (int c = 0; c < HID; ++c) {
            const float v = hr[c];
            o0 += v * w[c * 2];
            o1 += v * w[c * 2 + 1];
        }
    }
    out[(long)node * 2]     = o0;
    out[(long)node * 2 + 1] = o1;
}

// ---------- host orchestration ----------
extern "C" void kernel_launch(void* const* d_in, const int* in_sizes, int n_in,
                              void* d_out, int out_size, void* d_ws, size_t ws_size,
                              hipStream_t stream) {
    const float*     x   = (const float*)d_in[0];
    const long long* ei  = (const long long*)d_in[1];
    const float*     W0  = (const float*)d_in[2];
    const float*     as0 = (const float*)d_in[3];
    const float*     ad0 = (const float*)d_in[4];
    const float*     b0  = (const float*)d_in[5];
    const float*     g0  = (const float*)d_in[6];
    const float*     be0 = (const float*)d_in[7];
    const float*     W1  = (const float*)d_in[8];
    const float*     as1 = (const float*)d_in[9];
    const float*     ad1 = (const float*)d_in[10];
    const float*     b1  = (const float*)d_in[11];
    const float*     g1  = (const float*)d_in[12];
    const float*     be1 = (const float*)d_in[13];
    const float*     Wc  = (const float*)d_in[14];
    const float*     bc  = (const float*)d_in[15];

    const int IN_CH   = 128;
    const int n_nodes = in_sizes[0] / IN_CH;
    const int n_edges = in_sizes[1] / 2;
    const int e_tot   = n_edges + n_nodes;   // + self loops

    // workspace carve-out
    char* p = (char*)d_ws;
    const size_t szH = (size_t)HEADS * n_nodes * HID * sizeof(float);
    const size_t szS = (size_t)HEADS * n_nodes * sizeof(float);
    float*    hbuf  = (float*)p;    p += szH;
    float*    agg   = (float*)p;    p += szH;
    float*    ssrc  = (float*)p;    p += szS;
    float*    sdst  = (float*)p;    p += szS;
    unsigned* menc  = (unsigned*)p; p += szS;
    float*    denom = (float*)p;    p += szS;
    float*    sums  = (float*)p;    p += HEADS * HID * sizeof(float);
    float*    sumsq = (float*)p;    p += HEADS * HID * sizeof(float);

    const int tiles      = (n_nodes + 15) / 16;
    const int gemm_blks  = (tiles + 7) / 8;          // 8 waves (tiles) per block
    const long nh_total  = (long)HEADS * n_nodes;
    const int  attn_blks = (int)((nh_total + 255) / 256);
    const long eh_total  = (long)e_tot * HEADS;
    const int  edge_blks = (int)((eh_total + 255) / 256);
    const int  aggr_blks = (e_tot + 7) / 8;          // 8 waves per 256-thread block
    const long el_total  = (long)HEADS * n_nodes * HID;
    const int  elem_blks = (int)((el_total + 255) / 256);

    const float* layer_in   = x;
    long         in_stride  = 0;       // x shared across heads
    int          Kdim       = 128;
    const float* Wl[2]  = { W0, W1 };
    const float* asl[2] = { as0, as1 };
    const float* adl[2] = { ad0, ad1 };
    const float* bl[2]  = { b0, b1 };
    const float* gl[2]  = { g0, g1 };
    const float* bel[2] = { be0, be1 };

    for (int layer = 0; layer < 2; ++layer) {
        gat_wmma_gemm<<<dim3(gemm_blks, HEADS), 256, 0, stream>>>(
            layer_in, in_stride, Kdim, Wl[layer], hbuf, n_nodes);
        attn_scores<<<attn_blks, 256, 0, stream>>>(hbuf, asl[layer], adl[layer],
                                                   ssrc, sdst, n_nodes);
        // zero the segment-softmax + aggregation state (menc/denom are adjacent)
        hipMemsetAsync(menc, 0, 2 * szS, stream);
        hipMemsetAsync(agg, 0, szH, stream);
        hipMemsetAsync(sums, 0, 2 * HEADS * HID * sizeof(float), stream);

        edge_max<<<edge_blks, 256, 0, stream>>>(ei, ssrc, sdst, menc,
                                                n_nodes, n_edges, e_tot);
        edge_denom<<<edge_blks, 256, 0, stream>>>(ei, ssrc, sdst, menc, denom,
                                                  n_nodes, n_edges, e_tot);
        edge_aggregate<<<aggr_blks, 256, 0, stream>>>(ei, ssrc, sdst, menc, denom,
                                                      hbuf, agg, n_nodes, n_edges, e_tot);
        bn_stats<<<dim3(64, HEADS), 256, 0, stream>>>(agg, bl[layer], sums, sumsq, n_nodes);
        bn_apply_elu<<<elem_blks, 256, 0, stream>>>(agg, bl[layer], sums, sumsq,
                                                    gl[layer], bel[layer], n_nodes);
        layer_in  = agg;                  // BN+ELU output feeds next layer
        in_stride = (long)n_nodes * HID;  // per-head features now
        Kdim      = HID;
    }

    classify<<<(n_nodes + 127) / 128, 128, 0, stream>>>(agg, Wc, bc, (float*)d_out, n_nodes);
}